// HeteroGAT_25426206392775
// MI455X (gfx1250) — compile-verified
//
#include <hip/hip_runtime.h>
#include <hip/hip_bf16.h>
#include <math.h>

// ---------------------------------------------------------------------------
// HeteroGAT on MI455X (gfx1250, wave32).
// Dense GEMMs run on V_WMMA_F32_16X16X4_F32 (full f32 precision matrix core).
// Edge softmax/scatter use native f32 global atomics.
// ---------------------------------------------------------------------------

#define N_USER 100000
#define N_ITEM 50000
#define NEDGE  400000
#define DU 128
#define DI 64
#define HID 64
#define NH 4
#define HC 256

typedef float v2f __attribute__((ext_vector_type(2)));
typedef float v8f __attribute__((ext_vector_type(8)));

// ---------------------------------------------------------------------------
// GEMM: C[M,N] = A[M,K] @ W[N,K]^T (+ bias[N] if bias != nullptr)
// One wave computes one 16x16 tile of C; 4 waves per block -> 64 cols/block.
// Requires M%16==0, N%64==0, K%4==0 (all shapes here satisfy this).
// A-frag (16x4 f32): lanes 0-15 hold M=lane, K={k0,k0+1}; lanes 16-31 hold
// M=lane-16, K={k0+2,k0+3}.  B-frag mirrors with N along lanes.
// C/D: VGPR j -> row m0+j (lanes 0-15) / m0+8+j (lanes 16-31), col = lane&15.
// ---------------------------------------------------------------------------
__global__ void gemm_wmma_f32(const float* __restrict__ A,
                              const float* __restrict__ W,
                              const float* __restrict__ bias,
                              float* __restrict__ C,
                              int M, int N, int K) {
  const int lane = threadIdx.x & 31;
  const int wave = threadIdx.x >> 5;
  const int l    = lane & 15;
  const int half = lane >> 4;            // 0: K 0..1, 1: K 2..3 within a step
  const int m0   = blockIdx.x * 16;
  const int n0   = (blockIdx.y * 4 + wave) * 16;

  const float* __restrict__ Arow = A + (size_t)(m0 + l) * K;
  const float* __restrict__ Wrow = W + (size_t)(n0 + l) * K;

  v8f acc = {};
#pragma unroll 4
  for (int k0 = 0; k0 < K; k0 += 4) {
    const int kb = k0 + 2 * half;
    v2f a, b;
    a.x = Arow[kb];  a.y = Arow[kb + 1];
    b.x = Wrow[kb];  b.y = Wrow[kb + 1];
    acc = __builtin_amdgcn_wmma_f32_16x16x4_f32(
        /*neg_a=*/false, a, /*neg_b=*/false, b,
        /*c_mod=*/(short)0, acc, /*reuse_a=*/false, /*reuse_b=*/false);
  }

  const int n = n0 + l;
  const float bn = bias ? bias[n] : 0.0f;
#pragma unroll
  for (int j = 0; j < 8; ++j) {
    const int m = m0 + j + 8 * half;
    C[(size_t)m * N + n] = acc[j] + bn;
  }
}

// ---------------------------------------------------------------------------
// al[n,h] = sum_c X[n, h*64+c] * att[h*64+c]
// One wave per node; lane covers 8 contiguous channels (all in one head),
// reduce over the 8 lanes of each head group via shuffles.
// ---------------------------------------------------------------------------
__global__ void att_logits(const float* __restrict__ X,
                           const float* __restrict__ att,
                           float* __restrict__ al, int Nn) {
  const int lane = threadIdx.x & 31;
  const int wave = threadIdx.x >> 5;
  const int node = blockIdx.x * (blockDim.x >> 5) + wave;
  if (node >= Nn) return;
  const float* xr = X + (size_t)node * HC + lane * 8;
  const float* ar = att + lane * 8;
  float s = 0.f;
#pragma unroll
  for (int j = 0; j < 8; ++j) s += xr[j] * ar[j];
  s += __shfl_xor(s, 1, 32);
  s += __shfl_xor(s, 2, 32);
  s += __shfl_xor(s, 4, 32);
  if ((lane & 7) == 0) al[(size_t)node * NH + (lane >> 3)] = s;
}

// ---------------------------------------------------------------------------
__global__ void init_f32(float* __restrict__ p, float v, size_t n) {
  size_t i = (size_t)blockIdx.x * blockDim.x + threadIdx.x;
  size_t stride = (size_t)gridDim.x * blockDim.x;
  for (; i < n; i += stride) p[i] = v;
}

// e[i,h] = leaky_relu(al_s[src]+al_d[dst], 0.2); atomic max into m[dst,h]
__global__ void edge_logits_max(const int* __restrict__ src,
                                const int* __restrict__ dst,
                                const float* __restrict__ al_s,
                                const float* __restrict__ al_d,
                                float* __restrict__ e,
                                float* __restrict__ m, int ne) {
  const int i = blockIdx.x * blockDim.x + threadIdx.x;
  if (i >= ne) return;
  const int s = src[i], d = dst[i];
#pragma unroll
  for (int h = 0; h < NH; ++h) {
    float v = al_s[(size_t)s * NH + h] + al_d[(size_t)d * NH + h];
    v = v > 0.f ? v : 0.2f * v;
    e[(size_t)i * NH + h] = v;
    atomicMax(&m[(size_t)d * NH + h], v);   // global_atomic_max_num_f32
  }
}

// e[i,h] <- exp(e[i,h] - m[dst,h]); den[dst,h] += e[i,h]
__global__ void edge_exp_sum(const int* __restrict__ dst,
                             const float* __restrict__ m,
                             float* __restrict__ e,
                             float* __restrict__ den, int ne) {
  const int i = blockIdx.x * blockDim.x + threadIdx.x;
  if (i >= ne) return;
  const int d = dst[i];
#pragma unroll
  for (int h = 0; h < NH; ++h) {
    const float ex = __expf(e[(size_t)i * NH + h] - m[(size_t)d * NH + h]);
    e[(size_t)i * NH + h] = ex;
    atomicAdd(&den[(size_t)d * NH + h], ex);
  }
}

// out[dst, :] += xs[src, :] * alpha[head]; one wave per edge, 8 floats/lane.
__global__ void edge_scatter(const int* __restrict__ src,
                             const int* __restrict__ dst,
                             const float* __restrict__ xs,
                             const float* __restrict__ ex,
                             const float* __restrict__ den,
                             float* __restrict__ out, int ne) {
  const int lane = threadIdx.x & 31;
  const int wave = threadIdx.x >> 5;
  const int i = blockIdx.x * (blockDim.x >> 5) + wave;
  if (i >= ne) return;
  const int s = src[i], d = dst[i];
  const int h = lane >> 3;
  const float alpha = ex[(size_t)i * NH + h] /
                      (den[(size_t)d * NH + h] + 1e-16f);
  const float* xr = xs + (size_t)s * HC + lane * 8;
  float* orow = out + (size_t)d * HC + lane * 8;
#pragma unroll
  for (int j = 0; j < 8; ++j)
    atomicAdd(&orow[j], xr[j] * alpha);     // global_atomic_add_f32
}

// out[idx] = elu(out[idx] + bias[idx % 256])
__global__ void bias_elu(float* __restrict__ out,
                         const float* __restrict__ bias, size_t n) {
  size_t i = (size_t)blockIdx.x * blockDim.x + threadIdx.x;
  size_t stride = (size_t)gridDim.x * blockDim.x;
  for (; i < n; i += stride) {
    float v = out[i] + bias[i & (HC - 1)];
    out[i] = v > 0.f ? v : expm1f(v);
  }
}

// ---------------------------------------------------------------------------
extern "C" void kernel_launch(void* const* d_in, const int* in_sizes, int n_in,
                              void* d_out, int out_size, void* d_ws, size_t ws_size,
                              hipStream_t stream) {
  const float* x_user      = (const float*)d_in[0];
  const float* x_item      = (const float*)d_in[1];
  const int*   ei_ui_src   = (const int*)d_in[2];
  const int*   ei_ui_dst   = (const int*)d_in[3];
  const int*   ei_iu_src   = (const int*)d_in[4];
  const int*   ei_iu_dst   = (const int*)d_in[5];
  const float* proj_user_w = (const float*)d_in[6];
  const float* proj_user_b = (const float*)d_in[7];
  const float* proj_item_w = (const float*)d_in[8];
  const float* proj_item_b = (const float*)d_in[9];
  const float* ui_lin_src_w = (const float*)d_in[10];
  const float* ui_lin_dst_w = (const float*)d_in[11];
  const float* ui_att_src   = (const float*)d_in[12];
  const float* ui_att_dst   = (const float*)d_in[13];
  const float* ui_bias      = (const float*)d_in[14];
  const float* iu_lin_src_w = (const float*)d_in[15];
  const float* iu_lin_dst_w = (const float*)d_in[16];
  const float* iu_att_src   = (const float*)d_in[17];
  const float* iu_att_dst   = (const float*)d_in[18];
  const float* iu_bias      = (const float*)d_in[19];
  float* out = (float*)d_out;

  // Workspace layout (floats). Directions reuse xs/xd/al/e/m/den sequentially.
  float* ws  = (float*)d_ws;
  float* hu  = ws;                          // 100000*64  = 6,400,000
  float* hi  = hu  + (size_t)N_USER * HID;  //  50000*64  = 3,200,000
  float* xs  = hi  + (size_t)N_ITEM * HID;  // up to 100000*256 = 25,600,000
  float* xd  = xs  + (size_t)N_USER * HC;   // up to 25,600,000
  float* als = xd  + (size_t)N_USER * HC;   // up to 400,000
  float* ald = als + (size_t)N_USER * NH;   // up to 400,000
  float* ebf = ald + (size_t)N_USER * NH;   // 400000*4 = 1,600,000
  float* mbf = ebf + (size_t)NEDGE * NH;    // up to 400,000
  float* den = mbf + (size_t)N_USER * NH;   // up to 400,000

  float* out_user = out;                          // rows 0 .. N_USER-1
  float* out_item = out + (size_t)N_USER * HC;    // rows N_USER .. end

  // 0) zero the accumulation target
  init_f32<<<4096, 256, 0, stream>>>(out, 0.0f, (size_t)(N_USER + N_ITEM) * HC);

  // 1) node projections (f32 WMMA GEMMs)
  gemm_wmma_f32<<<dim3(N_USER / 16, DU >= 0 ? 1 : 1), 128, 0, stream>>>(
      x_user, proj_user_w, proj_user_b, hu, N_USER, HID, DU);
  gemm_wmma_f32<<<dim3(N_ITEM / 16, 1), 128, 0, stream>>>(
      x_item, proj_item_w, proj_item_b, hi, N_ITEM, HID, DI);

  // ------------------- direction 1: user -> item (dst = item) --------------
  gemm_wmma_f32<<<dim3(N_USER / 16, HC / 64), 128, 0, stream>>>(
      hu, ui_lin_src_w, nullptr, xs, N_USER, HC, HID);
  gemm_wmma_f32<<<dim3(N_ITEM / 16, HC / 64), 128, 0, stream>>>(
      hi, ui_lin_dst_w, nullptr, xd, N_ITEM, HC, HID);

  att_logits<<<(N_USER + 7) / 8, 256, 0, stream>>>(xs, ui_att_src, als, N_USER);
  att_logits<<<(N_ITEM + 7) / 8, 256, 0, stream>>>(xd, ui_att_dst, ald, N_ITEM);

  init_f32<<<256, 256, 0, stream>>>(mbf, -1e30f, (size_t)N_ITEM * NH);
  init_f32<<<256, 256, 0, stream>>>(den, 0.0f,   (size_t)N_ITEM * NH);

  edge_logits_max<<<(NEDGE + 255) / 256, 256, 0, stream>>>(
      ei_ui_src, ei_ui_dst, als, ald, ebf, mbf, NEDGE);
  edge_exp_sum<<<(NEDGE + 255) / 256, 256, 0, stream>>>(
      ei_ui_dst, mbf, ebf, den, NEDGE);
  edge_scatter<<<(NEDGE + 7) / 8, 256, 0, stream>>>(
      ei_ui_src, ei_ui_dst, xs, ebf, den, out_item, NEDGE);

  // ------------------- direction 2: item -> user (dst = user) --------------
  gemm_wmma_f32<<<dim3(N_ITEM / 16, HC / 64), 128, 0, stream>>>(
      hi, iu_lin_src_w, nullptr, xs, N_ITEM, HC, HID);
  gemm_wmma_f32<<<dim3(N_USER / 16, HC / 64), 128, 0, stream>>>(
      hu, iu_lin_dst_w, nullptr, xd, N_USER, HC, HID);

  att_logits<<<(N_ITEM + 7) / 8, 256, 0, stream>>>(xs, iu_att_src, als, N_ITEM);
  att_logits<<<(N_USER + 7) / 8, 256, 0, stream>>>(xd, iu_att_dst, ald, N_USER);

  init_f32<<<256, 256, 0, stream>>>(mbf, -1e30f, (size_t)N_USER * NH);
  init_f32<<<256, 256, 0, stream>>>(den, 0.0f,   (size_t)N_USER * NH);

  edge_logits_max<<<(NEDGE + 255) / 256, 256, 0, stream>>>(
      ei_iu_src, ei_iu_dst, als, ald, ebf, mbf, NEDGE);
  edge_exp_sum<<<(NEDGE + 255) / 256, 256, 0, stream>>>(
      ei_iu_dst, mbf, ebf, den, NEDGE);
  edge_scatter<<<(NEDGE + 7) / 8, 256, 0, stream>>>(
      ei_iu_src, ei_iu_dst, xs, ebf, den, out_user, NEDGE);

  // 3) bias + ELU epilogues (user rows get iu_bias, item rows get ui_bias)
  bias_elu<<<4096, 256, 0, stream>>>(out_user, iu_bias, (size_t)N_USER * HC);
  bias_elu<<<4096, 256, 0, stream>>>(out_item, ui_bias, (size_t)N_ITEM * HC);
}